// Attention_2_59373627899920
// MI455X (gfx1250) — compile-verified
//
#include <hip/hip_runtime.h>
#include <math.h>

#define NHEAD 6
#define HD    32
#define NTOK  256
#define NWIN  64
#define BATCH 256

typedef float v2f __attribute__((ext_vector_type(2)));
typedef float v8f __attribute__((ext_vector_type(8)));

#if __has_builtin(__builtin_amdgcn_sqrtf)
#define SQRTF(x) __builtin_amdgcn_sqrtf(x)
#else
#define SQRTF(x) sqrtf(x)
#endif
#if __has_builtin(__builtin_amdgcn_exp2f)
#define EXP2F(x) __builtin_amdgcn_exp2f(x)
#else
#define EXP2F(x) exp2f(x)
#endif

#define LOG2E 1.44269504088896340736f

__device__ __forceinline__ v8f wmma_f32_16x16x4(v2f a, v2f b, v8f c) {
  // D = A(16x4) x B(4x16) + C, fp32 in / fp32 acc  ->  v_wmma_f32_16x16x4_f32
  return __builtin_amdgcn_wmma_f32_16x16x4_f32(false, a, false, b,
                                               (short)0, c, false, false);
}

// gfx1250 async global->LDS copies (ASYNCcnt-tracked); lds addr = low 32 bits
// of the generic shared pointer (wave-relative LDS offset per aperture rules).
__device__ __forceinline__ void async_g2l_b128(unsigned lds, unsigned long long ga) {
  asm volatile("global_load_async_to_lds_b128 %0, %1, off"
               :: "v"(lds), "v"(ga) : "memory");
}
__device__ __forceinline__ void async_g2l_b32(unsigned lds, unsigned long long ga) {
  asm volatile("global_load_async_to_lds_b32 %0, %1, off"
               :: "v"(lds), "v"(ga) : "memory");
}
__device__ __forceinline__ void wait_async0() {
  asm volatile("s_wait_asynccnt 0x0" ::: "memory");
}
// Wave-local ordering for the wave-private Pl round trip: same-wave LDS ops
// complete in order (DScnt in-order per ISA), so we only need a compiler
// scheduling/memory barrier -- no s_barrier, no hardware instruction.
__device__ __forceinline__ void wave_lds_fence() {
  asm volatile("" ::: "memory");
  __builtin_amdgcn_wave_barrier();
  asm volatile("" ::: "memory");
}

__global__ __launch_bounds__(256)
void swin_euclid_attn_kernel(const float* __restrict__ qg,
                             const float* __restrict__ kg,
                             const float* __restrict__ vg,
                             const float* __restrict__ table,
                             const float* __restrict__ mask,
                             const int*   __restrict__ index,
                             float* __restrict__ outp)
{
  __shared__ float Qs[NTOK][HD];        // 32 KB, row major
  __shared__ float Ks[NTOK][HD];        // 32 KB, row major
  __shared__ float Vt[HD][NTOK + 2];    // 33 KB, transposed (padded, even stride)
  __shared__ float q2s[NTOK];
  __shared__ float k2s[NTOK];
  __shared__ float Pl[8][16][20];       // per-wave P-tile buffer (16B-aligned rows)

  const int tid   = threadIdx.x;
  const int wid   = tid >> 5;           // wave32
  const int lane  = tid & 31;
  const int lmod  = lane & 15;
  const int lhalf = lane >> 4;

  const int bh  = blockIdx.x;           // 0 .. B*NH-1
  const int b   = bh / NHEAD;
  const int h   = bh - b * NHEAD;
  const int wdx = b & (NWIN - 1);       // window index = b % 64

  // ---- async-stage Q, K (row major) and V (transposed via per-lane LDS dest) ----
  const size_t base = (size_t)bh * NTOK * HD;
  {
    const float* qp = qg + base;
    const float* kp = kg + base;
    for (int i = tid; i < NTOK * HD / 4; i += 256) {   // 8 x b128 per thread
      async_g2l_b128((unsigned)(uintptr_t)&Qs[0][i * 4],
                     (unsigned long long)(uintptr_t)(qp + i * 4));
      async_g2l_b128((unsigned)(uintptr_t)&Ks[0][i * 4],
                     (unsigned long long)(uintptr_t)(kp + i * 4));
    }
    const float* vp = vg + base;
    for (int e = tid; e < NTOK * HD; e += 256) {       // transpose on the fly
      int n = e >> 5, d = e & (HD - 1);
      async_g2l_b32((unsigned)(uintptr_t)&Vt[d][n],
                    (unsigned long long)(uintptr_t)(vp + e));
    }
    wait_async0();
  }
  __syncthreads();

  // ---- squared norms ----
  {
    float sq = 0.f, sk = 0.f;
    #pragma unroll
    for (int d = 0; d < HD; ++d) {
      float a = Qs[tid][d]; sq += a * a;
      float c = Ks[tid][d]; sk += c * c;
    }
    q2s[tid] = sq;
    k2s[tid] = sk;
  }
  __syncthreads();

  const float* maskw = mask + (size_t)wdx * NTOK * NTOK;

  for (int rti = 0; rti < 2; ++rti) {
    const int rt = rti * 8 + wid;          // row tile (16 rows), 16 tiles / 8 waves
    const int i  = rt * 16 + lmod;         // attention row owned by this lane
    const float q2i = q2s[i];
    const int*   irow = index + i * NTOK;
    const float* mrow = maskw + i * NTOK;

    // Q fragment = B operand of S^T = K x Q^T; fixed for this row tile
    v2f qfrag[8];
    #pragma unroll
    for (int s = 0; s < 8; ++s)
      qfrag[s] = *(const v2f*)&Qs[i][s * 4 + lhalf * 2];

    // ===== pass A: scores (base-2 scaled) cached in VGPRs; max-only reduction ====
    v8f valc[16];                          // 128 VGPRs: this lane's score cache
    float m2 = -3.0e38f;
    #pragma unroll
    for (int jt = 0; jt < 16; ++jt) {
      const int jb = jt * 16;
      v8f acc = {0.f,0.f,0.f,0.f,0.f,0.f,0.f,0.f};
      #pragma unroll
      for (int s = 0; s < 8; ++s) {
        v2f kfrag = *(const v2f*)&Ks[jb + lmod][s * 4 + lhalf * 2];
        acc = wmma_f32_16x16x4(kfrag, qfrag[s], acc);   // S^T tile
      }
      #pragma unroll
      for (int p = 0; p < 8; ++p) {
        const int j = jb + p + lhalf * 8;               // D layout: M = p + 8*half
        float s2  = q2i + k2s[j] - 2.0f * acc[p];
        float val = SQRTF(fmaxf(s2, 0.0f))
                  + table[irow[j] * NHEAD + h] + mrow[j];
        float v2  = val * LOG2E;                         // base-2 domain
        valc[jt][p] = v2;
        m2 = fmaxf(m2, v2);
      }
    }
    m2 = fmaxf(m2, __shfl_xor(m2, 16, 32));  // merge half-wave partials for row i

    // ===== pass B: e = 2^(v - m); accumulate row sum; out += P @ V (unnormalized) =
    float lr = 0.0f;
    v8f o0 = {0.f,0.f,0.f,0.f,0.f,0.f,0.f,0.f};
    v8f o1 = {0.f,0.f,0.f,0.f,0.f,0.f,0.f,0.f};
    #pragma unroll
    for (int jt = 0; jt < 16; ++jt) {
      const int jb = jt * 16;
      float e[8];
      #pragma unroll
      for (int p = 0; p < 8; ++p)
        e[p] = EXP2F(valc[jt][p] - m2);                  // single v_exp_f32 each
      // tree-reduce the tile's contribution to the row sum (no serial dep chain)
      lr += ((e[0] + e[1]) + (e[2] + e[3])) + ((e[4] + e[5]) + (e[6] + e[7]));
      // P^T -> transposed store: lane's 8 values are contiguous, two b128 stores
      {
        float4 lo = {e[0], e[1], e[2], e[3]};
        float4 hi = {e[4], e[5], e[6], e[7]};
        float4* dst = (float4*)&Pl[wid][lmod][lhalf * 8];
        dst[0] = lo;
        dst[1] = hi;
      }
      wave_lds_fence();  // wave-private buffer: same-wave LDS in-order suffices
      #pragma unroll
      for (int s = 0; s < 4; ++s) {       // K = 16 columns of P -> 4 WMMA steps
        const int kk = s * 4 + lhalf * 2;
        v2f pa = *(const v2f*)&Pl[wid][lmod][kk];       // A: P(i, j)
        v2f b0 = *(const v2f*)&Vt[lmod][jb + kk];       // B: V(j, d), d = lmod
        v2f b1 = *(const v2f*)&Vt[16 + lmod][jb + kk];  // B: d = 16 + lmod
        o0 = wmma_f32_16x16x4(pa, b0, o0);
        o1 = wmma_f32_16x16x4(pa, b1, o1);
      }
      wave_lds_fence();  // order PV reads before next tile's Pl overwrite (WAR)
    }
    lr += __shfl_xor(lr, 16, 32);          // full row sum (both halves)
    const float rL = 1.0f / lr;

    // ---- store with deferred normalization: row r stats live at lane r%16 ----
    float* orow = outp + (size_t)b * NTOK * (NHEAD * HD) + h * HD;
    #pragma unroll
    for (int p = 0; p < 8; ++p) {
      const int r16 = p + lhalf * 8;                 // row % 16 for this accum VGPR
      const float rp = __shfl(rL, r16, 32);          // broadcast that row's 1/L
      const int row = rt * 16 + r16;
      float* op = orow + (size_t)row * (NHEAD * HD);
      op[lmod]      = o0[p] * rp;
      op[16 + lmod] = o1[p] * rp;
    }
  }
}

extern "C" void kernel_launch(void* const* d_in, const int* in_sizes, int n_in,
                              void* d_out, int out_size, void* d_ws, size_t ws_size,
                              hipStream_t stream) {
  (void)in_sizes; (void)n_in; (void)out_size; (void)d_ws; (void)ws_size;
  const float* q     = (const float*)d_in[0];
  const float* k     = (const float*)d_in[1];
  const float* v     = (const float*)d_in[2];
  const float* table = (const float*)d_in[3];
  const float* mask  = (const float*)d_in[4];
  const int*   index = (const int*)d_in[5];
  float* out = (float*)d_out;

  dim3 grid(BATCH * NHEAD);   // 1536 workgroups, one per (batch-window, head)
  dim3 block(256);            // 8 wave32 waves
  swin_euclid_attn_kernel<<<grid, block, 0, stream>>>(q, k, v, table, mask, index, out);
}